// DPSH_D_Loss_88167088652900
// MI455X (gfx1250) — compile-verified
//
#include <hip/hip_runtime.h>
#include <math.h>

typedef __attribute__((ext_vector_type(2))) float v2f;
typedef __attribute__((ext_vector_type(8))) float v8f;

#define N_BS     128
#define BIT      48
#define N_CLS    100
#define CBIT     16
#define N_TRAIN  10000
#define N_TILES  (N_TRAIN / 16)   // 625
#define GRID_Y   40

// Raw hardware transcendentals: v_exp_f32 computes 2^x, v_log_f32 computes
// log2(x). Our log argument is always in [1,2], so the OCML denorm-rescue
// sequences (__logf) are pure overhead in the hot loop.
#if defined(__has_builtin)
#if __has_builtin(__builtin_amdgcn_exp2f) && __has_builtin(__builtin_amdgcn_logf)
#define FAST_EXP2(x) __builtin_amdgcn_exp2f(x)
#define FAST_LOG2(x) __builtin_amdgcn_logf(x)
#endif
#endif
#ifndef FAST_EXP2
#define FAST_EXP2(x) __expf((x) * 0.6931471805599453f)
#define FAST_LOG2(x) (__logf(x) * 1.4426950408889634f)
#endif

#define LOG2E 1.4426950408889634f
#define LN2   0.6931471805599453f

// ---------------------------------------------------------------------------
// Kernel 0: zero the two scalar accumulators and build the row-replacement
// map (map[t] = b if ind[b] == t else -1), implementing Y[ind]=y / U[ind]=u_
// without mutating the input buffers.
// ---------------------------------------------------------------------------
__global__ void k_init(const int* __restrict__ ind, int* __restrict__ map,
                       float* __restrict__ acc) {
    int t = blockIdx.x * blockDim.x + threadIdx.x;
    if (t == 0) { acc[0] = 0.0f; acc[1] = 0.0f; }
    if (t < N_TRAIN) {
        int m = -1;
#pragma unroll 8
        for (int b = 0; b < N_BS; ++b)
            if (ind[b] == t) m = b;
        map[t] = m;
    }
}

// ---------------------------------------------------------------------------
// Kernel 1: u_[b][c][d] = sum_k u[b][k] * w_D[c][k][d]   (one block per class)
// Also accumulates the quantization partial sum((x - sign(x))^2).
// ---------------------------------------------------------------------------
__global__ void __launch_bounds__(256) k_uproj(const float* __restrict__ u,
                                               const float* __restrict__ wD,
                                               float* __restrict__ u_ws,
                                               float* __restrict__ acc) {
    const int c = blockIdx.x;
    __shared__ float wd[BIT * CBIT];   // 48 x 16
    __shared__ float red[256];
    for (int i = threadIdx.x; i < BIT * CBIT; i += 256)
        wd[i] = wD[c * (BIT * CBIT) + i];
    __syncthreads();

    const int b  = threadIdx.x >> 1;          // 0..127
    const int d0 = (threadIdx.x & 1) * 8;     // 0 or 8
    float a[8];
#pragma unroll
    for (int j = 0; j < 8; ++j) a[j] = 0.0f;
    const float* urow = u + b * BIT;
#pragma unroll 4
    for (int k = 0; k < BIT; ++k) {
        float uv = urow[k];
#pragma unroll
        for (int j = 0; j < 8; ++j)
            a[j] = fmaf(uv, wd[k * CBIT + d0 + j], a[j]);
    }
    float q = 0.0f;
#pragma unroll
    for (int j = 0; j < 8; ++j) {
        float x = a[j];
        u_ws[b * (N_CLS * CBIT) + c * CBIT + d0 + j] = x;
        float sgn = (x > 0.0f) ? 1.0f : ((x < 0.0f) ? -1.0f : 0.0f);
        float dq  = x - sgn;
        q = fmaf(dq, dq, q);
    }
    red[threadIdx.x] = q;
    __syncthreads();
    for (int s = 128; s > 0; s >>= 1) {
        if (threadIdx.x < s) red[threadIdx.x] += red[threadIdx.x + s];
        __syncthreads();
    }
    if (threadIdx.x == 0) atomicAdd(&acc[1], red[0]);
}

// ---------------------------------------------------------------------------
// Kernel 2: fused  ip = einsum('tcd,bcd->btc')  +  loss epilogue.
// grid = (100 classes, 40 persistent blocks); block = 256 threads (8 waves).
// Each wave owns one 16-row batch tile; per t-tile: 4x V_WMMA_F32_16X16X4_F32
// chained over K=16, then the elementwise loss on the 8 accumulator VGPRs.
// ---------------------------------------------------------------------------
__global__ void __launch_bounds__(256) k_loss(const float* __restrict__ u_ws,
                                              const float* __restrict__ U,
                                              const float* __restrict__ Yb,
                                              const float* __restrict__ y,
                                              const int* __restrict__ map,
                                              float* __restrict__ acc) {
    const int c = blockIdx.x;
    __shared__ float u_c[N_BS * 17];   // u_[:, c, :], rows padded to 17
    __shared__ float Ut[16 * 17];      // U_eff tile (t_local x d), padded
    __shared__ float Yc[16];           // Y_eff[t, c]
    __shared__ float yb[N_BS];         // y[:, c]
    __shared__ float red[256];

    const int tid = threadIdx.x;
    // stage A operand (invariant across all t-tiles of this block)
    for (int i = tid; i < N_BS * CBIT; i += 256) {
        int b = i >> 4, d = i & 15;
        u_c[b * 17 + d] = u_ws[b * (N_CLS * CBIT) + c * CBIT + d];
    }
    if (tid < N_BS) yb[tid] = y[tid * N_CLS + c];
    __syncthreads();

    const int lane  = tid & 31;
    const int wave  = tid >> 5;           // 0..7 -> batch tile
    const int lrow  = lane & 15;
    const int khalf = (lane >> 4) << 1;   // 0 or 2 (K-pair select)

    // A fragments for the 4 K-steps (ISA 32-bit A 16x4 layout)
    v2f afrag[4];
    {
        const int m = wave * 16 + lrow;
#pragma unroll
        for (int kk = 0; kk < 4; ++kk) {
            afrag[kk].x = u_c[m * 17 + kk * 4 + khalf];
            afrag[kk].y = u_c[m * 17 + kk * 4 + khalf + 1];
        }
    }

    float lacc = 0.0f;
    for (int tile = blockIdx.y; tile < N_TILES; tile += GRID_Y) {
        const int t0 = tile << 4;
        // cooperative load of U tile + Y slice with scatter-replacement
        {
            const int row = tid >> 4, dcol = tid & 15;
            const int t   = t0 + row;
            const int rep = map[t];
            float v = (rep >= 0)
                          ? u_ws[rep * (N_CLS * CBIT) + c * CBIT + dcol]
                          : U[t * (N_CLS * CBIT) + c * CBIT + dcol];
            Ut[row * 17 + dcol] = v;
            if (dcol == 0)
                Yc[row] = (rep >= 0) ? y[rep * N_CLS + c] : Yb[t * N_CLS + c];
        }
        // prefetch next tile's U rows into cache (global_prefetch_b8)
        {
            const int ntile = tile + GRID_Y;
            if (ntile < N_TILES) {
                const float* p =
                    &U[(ntile * 16 + (tid >> 4)) * (N_CLS * CBIT) + c * CBIT];
                __builtin_prefetch(p, 0, 0);
            }
        }
        __syncthreads();

        // ip tile: 16(b) x 16(t), K=16 via 4 chained f32 WMMAs
        v8f cacc = {0.f, 0.f, 0.f, 0.f, 0.f, 0.f, 0.f, 0.f};
#pragma unroll
        for (int kk = 0; kk < 4; ++kk) {
            v2f bfrag;
            bfrag.x = Ut[lrow * 17 + kk * 4 + khalf];
            bfrag.y = Ut[lrow * 17 + kk * 4 + khalf + 1];
            cacc = __builtin_amdgcn_wmma_f32_16x16x4_f32(
                false, afrag[kk], false, bfrag, (short)0, cacc, false, false);
        }

        // fused loss epilogue on the D-matrix registers
        // (C/D layout: lane%16 = N (t), VGPR r -> M = r + 8*(lane/16))
        // log1p(exp(-|x|)) = ln2 * log2(1 + 2^(-|x|*log2e)); arg of log2 is
        // in [1,2] so the raw v_log_f32 needs no denorm handling.
        const float Yt    = Yc[lrow];
        const int   bbase = wave * 16 + ((lane >> 4) << 3);
#pragma unroll
        for (int r = 0; r < 8; ++r) {
            float ipv = cacc[r];
            float sv  = yb[bbase + r] * Yt;
            float e   = FAST_EXP2(-fabsf(ipv) * LOG2E);
            float lg  = FAST_LOG2(1.0f + e) * LN2;
            lacc += lg + fmaxf(ipv, 0.0f) - sv * ipv;
        }
        __syncthreads();
    }

    red[tid] = lacc;
    __syncthreads();
    for (int s = 128; s > 0; s >>= 1) {
        if (tid < s) red[tid] += red[tid + s];
        __syncthreads();
    }
    if (tid == 0) atomicAdd(&acc[0], red[0]);
}

// ---------------------------------------------------------------------------
// Kernel 3: final scalar = mean(likelihood) + alpha * mean(quantization)
// ---------------------------------------------------------------------------
__global__ void k_final(const float* __restrict__ acc, float* __restrict__ out) {
    out[0] = acc[0] * (1.0f / (128.0f * 10000.0f * 100.0f)) +
             0.1f * acc[1] * (1.0f / (128.0f * 100.0f * 16.0f));
}

// ---------------------------------------------------------------------------
extern "C" void kernel_launch(void* const* d_in, const int* in_sizes, int n_in,
                              void* d_out, int out_size, void* d_ws, size_t ws_size,
                              hipStream_t stream) {
    const float* u   = (const float*)d_in[0];
    const float* y   = (const float*)d_in[1];
    const int*   ind = (const int*)d_in[2];
    const float* U   = (const float*)d_in[3];
    const float* Yb  = (const float*)d_in[4];
    const float* wD  = (const float*)d_in[5];
    float* out = (float*)d_out;

    // workspace layout
    float* acc  = (float*)d_ws;                              // 2 floats
    int*   map  = (int*)((char*)d_ws + 256);                 // 10000 ints
    float* u_ws = (float*)((char*)d_ws + 256 + 40192);       // 128*100*16 f32

    k_init<<<(N_TRAIN + 255) / 256, 256, 0, stream>>>(ind, map, acc);
    k_uproj<<<N_CLS, 256, 0, stream>>>(u, wD, u_ws, acc);
    k_loss<<<dim3(N_CLS, GRID_Y), 256, 0, stream>>>(u_ws, U, Yb, y, map, acc);
    k_final<<<1, 1, 0, stream>>>(acc, out);
}